// ToRGBLayer_43104291782963
// MI455X (gfx1250) — compile-verified
//
#include <hip/hip_runtime.h>
#include <hip/hip_bf16.h>

typedef __attribute__((ext_vector_type(2))) float v2f;
typedef __attribute__((ext_vector_type(8))) float v8f;

#define IN_CH   256
#define OUT_CH  3
#define W_DIM   512
#define SPATIAL (64 * 64 * 64)   // 262144
#define NBATCH  2
#define KSTEPS  64               // IN_CH / 4 channels per WMMA K-step

static __device__ __forceinline__ float clip256(float v) {
    return fminf(fmaxf(v, -256.0f), 256.0f);
}

// ---------------------------------------------------------------------------
// Prologue: styles = (w @ affine_weight^T) * AFFINE_GAIN + affine_bias, then
// scaled by WEIGHT_GAIN and folded into wk. Result written to workspace in
// the exact V_WMMA_F32_16X16X4_F32 A-operand lane layout:
//   Aws[b][k][lane][0] = M[b][lane&15][4k + 2*(lane>>4) + 0]   (VGPR0, K even)
//   Aws[b][k][lane][1] = M[b][lane&15][4k + 2*(lane>>4) + 1]   (VGPR1, K odd)
// Rows 3..15 are zero padding.
// ---------------------------------------------------------------------------
__global__ __launch_bounds__(256) void build_a_kernel(
    const float* __restrict__ w,        // (B, 512)
    const float* __restrict__ aw,       // (256, 512)
    const float* __restrict__ abias,    // (256,)
    const float* __restrict__ weight,   // (3, 256, 1, 1, 1)
    float* __restrict__ Aws)            // (B, 64, 32, 2)
{
    __shared__ float wsh[W_DIM];
    __shared__ float ss[IN_CH];
    const int b   = blockIdx.x;
    const int tid = threadIdx.x;

    for (int j = tid; j < W_DIM; j += 256) wsh[j] = w[b * W_DIM + j];
    __syncthreads();

    {   // one channel per thread (256 threads == IN_CH)
        const float* awr = aw + tid * W_DIM;
        float acc = 0.0f;
        #pragma unroll 4
        for (int j = 0; j < W_DIM; ++j) acc = fmaf(wsh[j], awr[j], acc);
        // AFFINE_GAIN = 1/sqrt(512), WEIGHT_GAIN = 1/sqrt(256)
        ss[tid] = (acc * 0.04419417382415922f + abias[tid]) * 0.0625f;
    }
    __syncthreads();

    float* Ab = Aws + b * (KSTEPS * 32 * 2);
    for (int idx = tid; idx < KSTEPS * 32 * 2; idx += 256) {
        const int k     = idx >> 6;
        const int r     = idx & 63;
        const int lane  = r >> 1;
        const int which = r & 1;
        const int row   = lane & 15;      // output channel (A-matrix row)
        const int sel   = lane >> 4;      // K half selector
        const int ch    = 4 * k + 2 * sel + which;
        float v = 0.0f;
        if (row < OUT_CH) v = weight[row * IN_CH + ch] * ss[ch];
        Ab[idx] = v;
    }
}

// ---------------------------------------------------------------------------
// Main: out[b] = M[b](16x256, padded) @ x[b](256x262144), bias + clip.
// One wave per 64 contiguous spatial positions (4 N-tiles of 16).
// B-operand layout for 16x16x4 f32: lanes 0-15 hold K rows {2sel,2sel+1}=... :
//   vgpr0 = x[ch=4k+2*sel ][s0 + n*16 + (lane&15)]
//   vgpr1 = x[ch=4k+2*sel+1][s0 + n*16 + (lane&15)]
// ---------------------------------------------------------------------------
__global__ __launch_bounds__(256) void torgb_main_kernel(
    const float* __restrict__ x,     // (B, 256, 262144)
    const float* __restrict__ Aws,   // (B, 64, 32, 2)
    const float* __restrict__ bias,  // (3,)
    float* __restrict__ out)         // (B, 3, 262144)
{
    const int lane = threadIdx.x & 31;
    const int wid  = (int)blockIdx.x * 8 + (threadIdx.x >> 5);  // 0..8191
    const int b    = wid >> 12;            // 4096 tiles per batch
    const int tile = wid & 4095;
    const long s0  = (long)tile * 64;

    const int nrow = lane & 15;
    const int sel  = lane >> 4;

    const float* xb = x + (long)b * IN_CH * SPATIAL;
    const float* ap = Aws + b * (KSTEPS * 32 * 2) + lane * 2;
    const float* p  = xb + (long)(2 * sel) * SPATIAL + s0 + nrow;

    v8f acc0 = {}, acc1 = {}, acc2 = {}, acc3 = {};

    for (int k = 0; k < KSTEPS; ++k) {
        v2f a = *(const v2f*)ap;
        ap += 64;

        // prefetch next K-step's channels (1 MiB channel stride defeats
        // linear prefetch); clamp on last iteration to a known-valid address
        const float* pf = (k < KSTEPS - 1) ? (p + 4l * SPATIAL) : p;
        __builtin_prefetch(pf, 0, 0);
        __builtin_prefetch(pf + SPATIAL, 0, 0);

        v2f b0, b1, b2, b3;
        b0.x = p[0];   b0.y = p[SPATIAL + 0];
        b1.x = p[16];  b1.y = p[SPATIAL + 16];
        b2.x = p[32];  b2.y = p[SPATIAL + 32];
        b3.x = p[48];  b3.y = p[SPATIAL + 48];

        acc0 = __builtin_amdgcn_wmma_f32_16x16x4_f32(false, a, false, b0, (short)0, acc0, false, false);
        acc1 = __builtin_amdgcn_wmma_f32_16x16x4_f32(false, a, false, b1, (short)0, acc1, false, false);
        acc2 = __builtin_amdgcn_wmma_f32_16x16x4_f32(false, a, false, b2, (short)0, acc2, false, false);
        acc3 = __builtin_amdgcn_wmma_f32_16x16x4_f32(false, a, false, b3, (short)0, acc3, false, false);

        p += 4l * SPATIAL;
    }

    // C/D layout: VGPR r, lanes 0-15 -> (M=r, N=lane). Rows 0..2 are the
    // real output channels; lanes 16-31 hold zero-padded rows 8..15.
    if (lane < 16) {
        float* ob = out + (long)b * OUT_CH * SPATIAL + s0 + lane;
        const float bv0 = bias[0], bv1 = bias[1], bv2 = bias[2];

        ob[0l * SPATIAL +  0] = clip256(acc0[0] + bv0);
        ob[1l * SPATIAL +  0] = clip256(acc0[1] + bv1);
        ob[2l * SPATIAL +  0] = clip256(acc0[2] + bv2);

        ob[0l * SPATIAL + 16] = clip256(acc1[0] + bv0);
        ob[1l * SPATIAL + 16] = clip256(acc1[1] + bv1);
        ob[2l * SPATIAL + 16] = clip256(acc1[2] + bv2);

        ob[0l * SPATIAL + 32] = clip256(acc2[0] + bv0);
        ob[1l * SPATIAL + 32] = clip256(acc2[1] + bv1);
        ob[2l * SPATIAL + 32] = clip256(acc2[2] + bv2);

        ob[0l * SPATIAL + 48] = clip256(acc3[0] + bv0);
        ob[1l * SPATIAL + 48] = clip256(acc3[1] + bv1);
        ob[2l * SPATIAL + 48] = clip256(acc3[2] + bv2);
    }
}

extern "C" void kernel_launch(void* const* d_in, const int* in_sizes, int n_in,
                              void* d_out, int out_size, void* d_ws, size_t ws_size,
                              hipStream_t stream) {
    const float* x      = (const float*)d_in[0];   // (2, 256, 64, 64, 64)
    const float* w      = (const float*)d_in[1];   // (2, 512)
    const float* aw     = (const float*)d_in[2];   // (256, 512)
    const float* abias  = (const float*)d_in[3];   // (256,)
    const float* weight = (const float*)d_in[4];   // (3, 256, 1, 1, 1)
    const float* bias   = (const float*)d_in[5];   // (3,)
    float* out          = (float*)d_out;           // (2, 3, 64, 64, 64)
    float* Aws          = (float*)d_ws;            // 2*64*32*2 floats = 32 KB

    build_a_kernel<<<NBATCH, 256, 0, stream>>>(w, aw, abias, weight, Aws);

    // 2 batches * 4096 tiles = 8192 waves, 8 waves (256 threads) per block
    torgb_main_kernel<<<1024, 256, 0, stream>>>(x, Aws, bias, out);
}